// ButterflyBlockDiagLayer_25769803813
// MI455X (gfx1250) — compile-verified
//
#include <hip/hip_runtime.h>
#include <stdint.h>

typedef __attribute__((ext_vector_type(2))) float v2f;
typedef __attribute__((ext_vector_type(8))) float v8f;
typedef __attribute__((ext_vector_type(4))) unsigned int v4u;
typedef __attribute__((ext_vector_type(4))) int v4i;
typedef __attribute__((ext_vector_type(8))) int v8i;

#define BATCH   8192
#define NDIM    4096
#define NBLK    64      // K = L = 64 blocks per stage
#define BLKD    64      // P = Q = R = S
#define ROWS    16      // batch rows per workgroup (WMMA M)
#define IPAD    65      // intermediate inner (per 64-col block) stride, dwords
#define ISTRIDE 4164    // intermediate row stride = 64*65 + 4  (== 4 mod 64)
#define WSTRIDE 68      // weight tile row stride in LDS, dwords (== 4 mod 64)

#define WT_FLOATS    (BLKD * WSTRIDE)          // 4352 per buffer
#define INTER_FLOATS (ROWS * ISTRIDE)          // 66624
#define RED_FLOATS   (4 * 32 * 8)              // 1024 (4 q-tiles x 32 lanes x 8 acc)
#define LDS_FLOATS   (INTER_FLOATS + 2 * WT_FLOATS + RED_FLOATS)  // 76352 -> 305,408 B

// ---- Tensor Data Mover: DMA one 64x64 fp32 tile global->LDS, inserting 4 pad
// dwords after every 64 dwords so the LDS tile lands with row stride WSTRIDE=68.
__device__ __forceinline__ void tdm_load_tile(const float* __restrict__ gsrc,
                                              unsigned lds_byte_off) {
    const unsigned long long ga = (unsigned long long)(uintptr_t)gsrc;
    v4u g0;
    g0[0] = 1u;                                   // count=1, user desc, no gather
    g0[1] = lds_byte_off;                         // LDS destination (bytes)
    g0[2] = (unsigned)(ga & 0xFFFFFFFFu);         // global_addr[31:0]
    g0[3] = (unsigned)((ga >> 32) & 0x01FFFFFFu)  // global_addr[56:32]
            | (2u << 30);                         // type = 2 ("image")
    v8i g1;
    g1[0] = (2 << 16)     // data_size = 4 bytes
          | (1 << 20)     // pad_enable
          | (5 << 22)     // pad_interval code 5 = every 64 dwords
          | (3 << 25);    // pad_amount   code 3 = 4 dwords
    g1[1] = (64 << 16);   // tensor_dim0[15:0] = 64 (atomic_barrier_addr = 0)
    g1[2] = (64 << 16);   // tensor_dim0 hi = 0; tensor_dim1[15:0] = 64
    g1[3] = (64 << 16);   // tensor_dim1 hi = 0; tile_dim0 = 64
    g1[4] = 64;           // tile_dim1 = 64; tile_dim2 = 0 (2D tile)
    g1[5] = 64;           // tensor_dim0_stride (lo32) = 64 elements
    g1[6] = 0;            // stride0 hi / stride1 lo
    g1[7] = 0;            // stride1 hi
    v4i gz = {0, 0, 0, 0};
#if defined(__clang_major__) && (__clang_major__ >= 23)
    v8i gz8 = {0, 0, 0, 0, 0, 0, 0, 0};
    __builtin_amdgcn_tensor_load_to_lds(g0, g1, gz, gz, gz8, 0);
#else
    __builtin_amdgcn_tensor_load_to_lds(g0, g1, gz, gz, 0);
#endif
}

__global__ __launch_bounds__(256)
void monarch_tdm_kernel(const float* __restrict__ x,
                        const float* __restrict__ w1,
                        const float* __restrict__ w2,
                        float* __restrict__ out) {
    extern __shared__ float smem[];
    float* inter = smem;                               // [16][ISTRIDE], phys m*ISTRIDE + l*IPAD + r
    float* wt    = smem + INTER_FLOATS;                // [2][64][WSTRIDE] TDM-filled tiles
    float* red   = smem + INTER_FLOATS + 2 * WT_FLOATS;// [4][8][32] partial-sum exchange

    const int tid  = threadIdx.x;
    const int lane = tid & 31;
    const int wave = tid >> 5;          // 0..7
    const int lm   = lane & 15;
    const int half = lane >> 4;         // K-pair select within a WMMA step
    const int qt   = wave & 3;          // 16-wide N tile
    const int kh   = wave >> 2;         // which half of the K=64 reduction
    const int b0   = blockIdx.x * ROWS;
    const unsigned wt_base = (unsigned)(INTER_FLOATS * sizeof(float));
    const unsigned wt_buf_bytes = (unsigned)(WT_FLOATS * sizeof(float));

    // ================= Stage 1: out1[b,k,q] = sum_p x[b,k,p] * w1[k,q,p] =================
    if (wave == 0) tdm_load_tile(w1, wt_base);                       // prologue: block 0
    for (int k = 0; k < NBLK; ++k) {
        const int cur = k & 1;
        if (wave == 0) __builtin_amdgcn_s_wait_tensorcnt(0);         // block k resident
        __syncthreads();                                             // publish tile; red free
        if (wave == 0 && (k + 1) < NBLK)                             // DMA next block while computing
            tdm_load_tile(w1 + (size_t)(k + 1) * (BLKD * BLKD),
                          wt_base + (unsigned)(cur ^ 1) * wt_buf_bytes);

        const float* wl   = wt + cur * WT_FLOATS;
        const float* xseg = x + (size_t)(b0 + lm) * NDIM + k * BLKD + kh * 32 + half * 2;
        const int    q    = qt * 16 + lm;

        v8f acc = {};
#pragma unroll
        for (int t = 0; t < 8; ++t) {
            v2f a = *reinterpret_cast<const v2f*>(xseg + t * 4);     // A(16x4): aligned 8B/lane
            const int p = kh * 32 + t * 4 + half * 2;
            v2f b;                                                    // B(p,q) = w1[k][q][p]
            b.x = wl[q * WSTRIDE + p];
            b.y = wl[q * WSTRIDE + p + 1];
            acc = __builtin_amdgcn_wmma_f32_16x16x4_f32(false, a, false, b,
                                                        (short)0, acc, false, false);
        }
        if (k + 1 < NBLK) __builtin_prefetch(xseg + BLKD, 0, 3);     // next x segment

        if (kh) {                                                    // upper-half partials -> LDS
#pragma unroll
            for (int v = 0; v < 8; ++v) red[qt * 256 + v * 32 + lane] = acc[v];
        }
        __syncthreads();
        if (!kh) {                                                   // combine + permuted scatter
#pragma unroll
            for (int v = 0; v < 8; ++v) acc[v] += red[qt * 256 + v * 32 + lane];
#pragma unroll
            for (int v = 0; v < 8; ++v) {
                const int m = v + half * 8;
                inter[m * ISTRIDE + q * IPAD + k] = acc[v];          // logical [m][l=q][r=k]
            }
        }
    }

    // ========= Stage 2: out2[b,l,s] = sum_r inter[b,l,r] * w2[l,s,r]; out[b, s*64+l] =========
    if (wave == 0) tdm_load_tile(w2, wt_base);                       // prologue: block 0
    for (int l = 0; l < NBLK; ++l) {
        const int cur = l & 1;
        if (wave == 0) __builtin_amdgcn_s_wait_tensorcnt(0);
        __syncthreads();                                             // also publishes last inter writes
        if (wave == 0 && (l + 1) < NBLK)
            tdm_load_tile(w2 + (size_t)(l + 1) * (BLKD * BLKD),
                          wt_base + (unsigned)(cur ^ 1) * wt_buf_bytes);

        const float* wl = wt + cur * WT_FLOATS;
        const int    s  = qt * 16 + lm;

        v8f acc = {};
#pragma unroll
        for (int t = 0; t < 8; ++t) {
            const int r0 = kh * 32 + t * 4 + half * 2;
            v2f a;                                                   // A from permuted intermediate
            a.x = inter[lm * ISTRIDE + l * IPAD + r0];
            a.y = inter[lm * ISTRIDE + l * IPAD + r0 + 1];
            v2f b;                                                   // B(r,s) = w2[l][s][r]
            b.x = wl[s * WSTRIDE + r0];
            b.y = wl[s * WSTRIDE + r0 + 1];
            acc = __builtin_amdgcn_wmma_f32_16x16x4_f32(false, a, false, b,
                                                        (short)0, acc, false, false);
        }

        if (kh) {
#pragma unroll
            for (int v = 0; v < 8; ++v) red[qt * 256 + v * 32 + lane] = acc[v];
        }
        __syncthreads();
        if (!kh) {
#pragma unroll
            for (int v = 0; v < 8; ++v) acc[v] += red[qt * 256 + v * 32 + lane];
#pragma unroll
            for (int v = 0; v < 8; ++v) {
                const int m = v + half * 8;
                out[(size_t)(b0 + m) * NDIM + s * BLKD + l] = acc[v]; // final permute folded in
            }
        }
    }
}

extern "C" void kernel_launch(void* const* d_in, const int* in_sizes, int n_in,
                              void* d_out, int out_size, void* d_ws, size_t ws_size,
                              hipStream_t stream) {
    (void)in_sizes; (void)n_in; (void)d_ws; (void)ws_size; (void)out_size;
    const float* x  = (const float*)d_in[0];
    const float* w1 = (const float*)d_in[1];
    const float* w2 = (const float*)d_in[2];
    float* out = (float*)d_out;

    const size_t lds_bytes = (size_t)LDS_FLOATS * sizeof(float);   // 305,408 B <= 320 KB
    (void)hipFuncSetAttribute((const void*)monarch_tdm_kernel,
                              hipFuncAttributeMaxDynamicSharedMemorySize,
                              (int)lds_bytes);

    const int grid = BATCH / ROWS;                                  // 512 workgroups
    monarch_tdm_kernel<<<grid, 256, lds_bytes, stream>>>(x, w1, w2, out);
}